// HelionGeluD8_45904610460294
// MI455X (gfx1250) — compile-verified
//
#include <hip/hip_runtime.h>
#include <hip/hip_bf16.h>

// D8 isotypic->regular -> GELU -> regular->isotypic, fully pointwise.
// Memory-bound streaming kernel: 512 MB traffic, ~22us floor at 23.3 TB/s.

typedef __attribute__((ext_vector_type(4))) float v4f;

#define KQ   0.3535533906f   // matches reference SQRT2_OVER_4
#define IS2  0.7071067812f   // matches reference SQRT_OVER_2

// Fast erf (Abramowitz-Stegun 7.1.26): |abs err| < 1.5e-7.
// ~11 FMA + 1 v_exp + 1 v_rcp instead of libm erff's much longer path,
// keeping ALU cost well below the HBM roofline.
__device__ __forceinline__ float gelu_fast(float x) {
    float u  = x * IS2;
    float au = fabsf(u);
    float t  = __builtin_amdgcn_rcpf(fmaf(0.3275911f, au, 1.0f));
    float p  = fmaf(t, 1.061405429f, -1.453152027f);
    p = fmaf(t, p, 1.421413741f);
    p = fmaf(t, p, -0.284496736f);
    p = fmaf(t, p, 0.254829592f);
    p = p * t;
    float e    = __expf(-au * au);          // v_exp_f32 (TRANS pipe)
    float erfa = fmaf(-p, e, 1.0f);         // erf(|u|)
    float erfu = copysignf(erfa, x);        // sign(u) == sign(x)
    return 0.5f * x * (1.0f + erfu);
}

__global__ __launch_bounds__(256) void HelionGeluD8_kernel(
    const v4f* __restrict__ xA1, const v4f* __restrict__ xA2,
    const v4f* __restrict__ xB1, const v4f* __restrict__ xB2,
    const v4f* __restrict__ x2d,
    v4f* __restrict__ oA1, v4f* __restrict__ oA2,
    v4f* __restrict__ oB1, v4f* __restrict__ oB2,
    v4f* __restrict__ o2d,
    int nvec)
{
    int tid = blockIdx.x * 256 + threadIdx.x;
    if (tid >= nvec) return;

    // C = 512 floats = 128 vec4 per (b,n) row for the A/B tensors.
    int c  = tid & 127;          // vec4 column within row
    int r  = tid >> 7;           // flattened (b,n) row, 0..16383
    int b2 = r * 512 + c;        // x_2d row = 2048 floats = 512 vec4

    // Streaming loads, non-temporal (no reuse anywhere; don't thrash L2).
    v4f X0 = __builtin_nontemporal_load(xA1 + tid);
    v4f X1 = __builtin_nontemporal_load(xA2 + tid);
    v4f X2 = __builtin_nontemporal_load(xB1 + tid);
    v4f X3 = __builtin_nontemporal_load(xB2 + tid);
    v4f Y0 = __builtin_nontemporal_load(x2d + b2);          // [:,:,0,:C]
    v4f Y1 = __builtin_nontemporal_load(x2d + b2 + 128);    // [:,:,0,C:]
    v4f Y2 = __builtin_nontemporal_load(x2d + b2 + 256);    // [:,:,1,:C]
    v4f Y3 = __builtin_nontemporal_load(x2d + b2 + 384);    // [:,:,1,C:]

    v4f O0 = {}, O1 = {}, O2 = {}, O3 = {}, O4 = {}, O5 = {}, O6 = {}, O7 = {};

#pragma unroll
    for (int j = 0; j < 4; ++j) {
        float x0 = X0[j], x1 = X1[j], x2 = X2[j], x3 = X3[j];
        float y0 = Y0[j], y1 = Y1[j], y2 = Y2[j], y3 = Y3[j];

        // isotypic_to_regular
        float a = x0 + x1, b = x0 - x1, cc = x2 + x3, d = x2 - x3;
        float e = y0 + y1, f = y0 - y1, g  = y2 + y3, h = y2 - y3;
        float apc = a + cc, amc = a - cc, bpd = b + d, bmd = b - d;
        float eph = e + h,  emh = e - h,  fpg = f + g, fmg = f - g;
        float r0 = KQ * (apc + eph), r1 = KQ * (amc + fmg);
        float r2 = KQ * (apc - eph), r3 = KQ * (amc - fmg);
        float r4 = KQ * (bpd - fpg), r5 = KQ * (bmd - emh);
        float r6 = KQ * (bpd + fpg), r7 = KQ * (bmd + emh);

        // GELU on all 8 regular components
        float g0 = gelu_fast(r0), g1 = gelu_fast(r1);
        float g2 = gelu_fast(r2), g3 = gelu_fast(r3);
        float g4 = gelu_fast(r4), g5 = gelu_fast(r5);
        float g6 = gelu_fast(r6), g7 = gelu_fast(r7);

        // regular_to_isotypic
        a = g0 + g1; b = g0 - g1; cc = g2 + g3; d = g2 - g3;
        e = g4 + g5; f = g4 - g5; g  = g6 + g7; h = g6 - g7;
        apc = a + cc; float cma = cc - a; bpd = b + d; bmd = b - d;
        float epg = e + g, gme = g - e, fph = f + h, fmh = f - h;

        O0[j] = KQ * (apc + epg);
        O1[j] = KQ * (apc - epg);
        O2[j] = KQ * (bpd + fph);
        O3[j] = KQ * (bpd - fph);
        O4[j] = KQ * (gme - cma);
        O5[j] = KQ * (bmd + fmh);
        O6[j] = KQ * (bmd - fmh);
        O7[j] = KQ * (gme + cma);
    }

    __builtin_nontemporal_store(O0, oA1 + tid);
    __builtin_nontemporal_store(O1, oA2 + tid);
    __builtin_nontemporal_store(O2, oB1 + tid);
    __builtin_nontemporal_store(O3, oB2 + tid);
    __builtin_nontemporal_store(O4, o2d + b2);          // y_2d[:,:,0,:C]
    __builtin_nontemporal_store(O5, o2d + b2 + 128);    // y_2d[:,:,0,C:]
    __builtin_nontemporal_store(O6, o2d + b2 + 256);    // y_2d[:,:,1,:C]
    __builtin_nontemporal_store(O7, o2d + b2 + 384);    // y_2d[:,:,1,C:]
}

extern "C" void kernel_launch(void* const* d_in, const int* in_sizes, int n_in,
                              void* d_out, int out_size, void* d_ws, size_t ws_size,
                              hipStream_t stream) {
    (void)n_in; (void)out_size; (void)d_ws; (void)ws_size;

    const v4f* xA1 = (const v4f*)d_in[0];
    const v4f* xA2 = (const v4f*)d_in[1];
    const v4f* xB1 = (const v4f*)d_in[2];
    const v4f* xB2 = (const v4f*)d_in[3];
    const v4f* x2d = (const v4f*)d_in[4];

    const int nA   = in_sizes[0];     // 16*1024*512 = 8388608 floats
    const int nvec = nA / 4;          // 2097152 vec4 work items

    float* out = (float*)d_out;
    v4f* oA1 = (v4f*)(out);                      // iso[0]
    v4f* oA2 = (v4f*)(out + (size_t)nA);         // iso[1]
    v4f* oB1 = (v4f*)(out + 2 * (size_t)nA);     // iso[2]
    v4f* oB2 = (v4f*)(out + 3 * (size_t)nA);     // iso[3]
    v4f* o2d = (v4f*)(out + 4 * (size_t)nA);     // y_2d (4*nA floats)

    dim3 block(256);
    dim3 grid((nvec + 255) / 256);
    HelionGeluD8_kernel<<<grid, block, 0, stream>>>(
        xA1, xA2, xB1, xB2, x2d, oA1, oA2, oB1, oB2, o2d, nvec);
}